// ADDSGNLM_76029511074252
// MI455X (gfx1250) — compile-verified
//
#include <hip/hip_runtime.h>
#include <cstddef>

// ---------------------------------------------------------------------------
// Guided non-local means (DSGNLM), 4 fused passes for MI455X / gfx1250.
//
// Per pass: for each 32x32 tile, stage guide (y) and blended-noisy tiles in
// LDS (reflect-mirrored), then for each of 81 window offsets compute the 7x7
// box-mean of the squared guide difference with TWO banded 16x16x4 f32 WMMA
// matmuls (horizontal then vertical 7-tap), weight w = exp(-d/sigma2(pixel)),
// and accumulate num/den in registers laid out as WMMA C/D fragments.
//
// CDNA5 specifics used:
//   * v_wmma_f32_16x16x4_f32 for both separable box passes (band matrices
//     are compile-time constant fragments; EXEC is all-ones at every WMMA).
//   * global_load_async_to_lds_b32 (+ s_wait_asynccnt) to stage the guide
//     tile, with reflect-mirrored per-lane addresses.
//
// Reflect-padding trick: Gd[local q] = y[mir(global q)], so the double mirror
// required by box_mean's reflect pad of the *difference field* is automatic:
//   sq_ext(p) = (Gd[m(p)] - Gd[m(p)+off])^2
// with m(p) folded into offset-invariant precomputed indices idx0; per offset
// only a scalar shift sh = oy*stride+ox is added. No branches in the hot loop.
//
// Fragment layout assumptions (wave32, from CDNA5 ISA 7.12.2):
//   A (16x4 f32): lane L -> M = L&15 ; comp c -> K = 2*(L>>4) + c
//   B (4x16 f32): lane L -> N = L&15 ; comp c -> K = 2*(L>>4) + c  (mirror of A)
//   C/D (16x16) : lane L -> N = L&15 ; vgpr v -> M = v + 8*(L>>4)
// ---------------------------------------------------------------------------

typedef float v2f __attribute__((ext_vector_type(2)));
typedef float v8f __attribute__((ext_vector_type(8)));

#define HH   1024
#define WW   1024
#define TILE 32

__device__ __forceinline__ int mir(int i, int n) {
  i = (i < 0) ? -i : i;                // reflect (no edge repeat)
  return (i >= n) ? (2 * n - 2 - i) : i;
}

__global__ __launch_bounds__(256, 2)
void dsgnlm_pass(const float* __restrict__ xin, const float* __restrict__ yg,
                 const float* __restrict__ hArr, int hIdx,
                 const float* __restrict__ sArr, int sIdx,
                 float* __restrict__ xout)
{
  // LDS: ~36 KB total
  __shared__ float Gd[46 * 47];       // guide tile, halo 7, odd stride (conflict-free)
  __shared__ float Nb[40 * 41];       // blended noisy tile, halo 4
  __shared__ float Hs[2][48 * 33];    // horizontal box sums, rows padded to 48
  __shared__ float RedN[32 * 33];
  __shared__ float RedD[32 * 33];

  const int tid  = threadIdx.x;
  const int wid  = tid >> 5;
  const int lane = tid & 31;
  const int hi   = (lane >> 4) & 1;
  const int nn   = lane & 15;

  const int tr0 = blockIdx.y * TILE;
  const int tc0 = blockIdx.x * TILE;
  const size_t ib = (size_t)blockIdx.z * HH * WW;

  const float hv   = hArr[hIdx];
  const float step = (sIdx >= 0) ? fminf(fmaxf(sArr[sIdx], 0.6f), 1.0f) : 1.0f;

  // ---- stage guide tile via CDNA5 async global->LDS copy ----
  for (int i = tid; i < 46 * 46; i += 256) {
    int r = i / 46, c = i % 46;
    int gr = mir(tr0 - 7 + r, HH);
    int gc = mir(tc0 - 7 + c, WW);
    const float* gp = &yg[ib + (size_t)gr * WW + gc];
    unsigned lds = (unsigned)(size_t)(&Gd[r * 47 + c]);   // low 32 bits = LDS offset
    asm volatile("global_load_async_to_lds_b32 %0, %1, off"
                 :: "v"(lds), "v"((unsigned long long)(size_t)gp) : "memory");
  }
  // ---- stage blended noisy tile: (1-step)*x + step*y, reflect-mirrored ----
  for (int i = tid; i < 40 * 40; i += 256) {
    int r = i / 40, c = i % 40;
    int gr = mir(tr0 - 4 + r, HH);
    int gc = mir(tc0 - 4 + c, WW);
    size_t g = ib + (size_t)gr * WW + gc;
    Nb[r * 41 + c] = (1.0f - step) * xin[g] + step * yg[g];
  }
  asm volatile("s_wait_asynccnt 0" ::: "memory");
  __syncthreads();

  // V-pass block owned by this wave (blocks 0..3 of the 32x32 tile)
  const int blk = wid & 3;
  const int rr = 16 * (blk >> 1) + 8 * hi;   // + v  -> output row in tile
  const int cc = 16 * (blk & 1) + nn;        //       output col in tile

  // Constant banded fragments: band(k,n) = (n <= k <= n+6). Identical formula
  // serves as B-matrix for the horizontal pass and A-matrix for the vertical.
  v2f bnd[6];
#pragma unroll
  for (int t = 0; t < 6; ++t) {
    int k0 = 4 * t + 2 * hi;
    bnd[t].x = (k0     >= nn && k0     <= nn + 6) ? 1.0f : 0.0f;
    bnd[t].y = (k0 + 1 >= nn && k0 + 1 <= nn + 6) ? 1.0f : 0.0f;
  }

  // Per-pixel sigma from Laplacian of the blended image: fold 1/49 and the
  // negation into one scale: w = exp(vsum * scl).
  float scl[8], num[8], den[8];
#pragma unroll
  for (int v = 0; v < 8; ++v) {
    int r = rr + v, c = cc;
    float lap = Nb[(r + 3) * 41 + (c + 4)] + Nb[(r + 5) * 41 + (c + 4)]
              + Nb[(r + 4) * 41 + (c + 3)] + Nb[(r + 4) * 41 + (c + 5)]
              - 4.0f * Nb[(r + 4) * 41 + (c + 4)];
    float sg = hv * lap;
    scl[v] = -1.0f / (49.0f * (sg * sg + 1e-8f));
    num[v] = 0.0f; den[v] = 0.0f;
  }

  // ---- per-wave H-pass block precomputation (offset-invariant) ----
  // Wave w owns H-blocks {w, w+8 (if <12)} of the 12 (buf,rowblk,colblk) blocks.
  int   idx0[2][12];   // mirrored local Gd index of each A-fragment element
  float g0v[2][12];    // center guide value (offset-invariant factor)
  int   sbase[2];      // Hs store base (elements)
  int   bufq[2];
  bool  valq[2];
#pragma unroll
  for (int q = 0; q < 2; ++q) {
    int qb = wid + 8 * q;
    valq[q] = (qb < 12);
    int qc = valq[q] ? qb : 0;
    int buf = qc / 6, rem = qc % 6, rblk = rem >> 1, cb = rem & 1;
    bufq[q]  = buf;
    sbase[q] = (rblk * 16 + 8 * hi) * 33 + 16 * cb + nn;
    int hsRow = rblk * 16 + nn;                  // A: M = row in Hs space
    int er = (hsRow > 37) ? 37 : hsRow;          // clamp: finite, band-zeroed
    int rowb = (mir(tr0 - 3 + er, HH) - tr0 + 7) * 47;
#pragma unroll
    for (int j = 0; j < 12; ++j) {
      int kc = 16 * cb + 4 * (j >> 1) + 2 * hi + (j & 1);  // A: K = sq column
      int ec = (kc > 37) ? 37 : kc;
      int ecm = mir(tc0 - 3 + ec, WW) - tc0 + 7;
      idx0[q][j] = rowb + ecm;
      g0v[q][j]  = Gd[rowb + ecm];
    }
  }
  // Phase-B (V-pass) offset-invariant bases
  const int hb0 = (16 * (blk >> 1) + 2 * hi) * 33 + cc;  // Hs read base
  const int nb0 = (rr + 4) * 41 + (cc + 4);              // Nb read base

  // ---- 81 offsets, processed two at a time (double-buffered Hs) ----
  for (int p = 0; p < 81; p += 2) {
    const bool validB = (p + 1 < 81);
    const int oA = p, oB = validB ? p + 1 : p;
    const int oyA = oA / 9 - 4, oxA = oA % 9 - 4;
    const int oyB = oB / 9 - 4, oxB = oB % 9 - 4;
    const int shA = oyA * 47 + oxA, shB = oyB * 47 + oxB;    // Gd shift
    const int snA = oyA * 41 + oxA, snB = oyB * 41 + oxB;    // Nb shift

    // Phase A: horizontal 7-tap sums Hs = SQ x Band; SQ A-fragments are
    // (g0 - Gd[idx0+sh])^2 computed on the fly.
#pragma unroll
    for (int q = 0; q < 2; ++q) {
      if (valq[q]) {                                // wave-uniform in practice
        const int sh = bufq[q] ? shB : shA;
        v8f acc = {0.f, 0.f, 0.f, 0.f, 0.f, 0.f, 0.f, 0.f};
#pragma unroll
        for (int t = 0; t < 6; ++t) {
          v2f a;
#pragma unroll
          for (int cmp = 0; cmp < 2; ++cmp) {
            float d = g0v[q][2 * t + cmp] - Gd[idx0[q][2 * t + cmp] + sh];
            a[cmp] = d * d;
          }
          acc = __builtin_amdgcn_wmma_f32_16x16x4_f32(false, a, false, bnd[t],
                                                      (short)0, acc, false, false);
        }
        float* hp = &Hs[bufq[q]][sbase[q]];
#pragma unroll
        for (int v = 0; v < 8; ++v) hp[v * 33] = acc[v];
      }
    }
    __syncthreads();

    // Phase B: vertical 7-tap via WMMA (D = Band x Hs), then weight+accum.
    // Waves 0-3 own offset A, waves 4-7 own offset B (skipped on odd tail).
    {
      const int buf = wid >> 2;
      if ((buf == 0) || validB) {                   // wave-uniform
        v8f acc = {0.f, 0.f, 0.f, 0.f, 0.f, 0.f, 0.f, 0.f};
        const float* hq = &Hs[buf][hb0];
#pragma unroll
        for (int t = 0; t < 6; ++t) {
          v2f bf;
          bf.x = hq[t * 132];                       // B: K = hsum row
          bf.y = hq[t * 132 + 33];
          acc = __builtin_amdgcn_wmma_f32_16x16x4_f32(false, bnd[t], false, bf,
                                                      (short)0, acc, false, false);
        }
        const float* nq = &Nb[nb0 + (buf ? snB : snA)];
#pragma unroll
        for (int v = 0; v < 8; ++v) {
          float wgt = __expf(acc[v] * scl[v]);
          float nv  = nq[v * 41];
          num[v] += wgt * nv;
          den[v] += wgt;
        }
      }
    }
    __syncthreads();
  }

  // ---- cross-wave (A/B halves) reduction and output ----
  if (wid >= 4) {
#pragma unroll
    for (int v = 0; v < 8; ++v) {
      RedN[(rr + v) * 33 + cc] = num[v];
      RedD[(rr + v) * 33 + cc] = den[v];
    }
  }
  __syncthreads();
  if (wid < 4) {
#pragma unroll
    for (int v = 0; v < 8; ++v) {
      float ns = num[v] + RedN[(rr + v) * 33 + cc];
      float ds = den[v] + RedD[(rr + v) * 33 + cc];
      xout[ib + (size_t)(tr0 + rr + v) * WW + (tc0 + cc)] = ns / ds;
    }
  }
}

// ---------------------------------------------------------------------------
// 4 passes ping-ponged through one 32 MB scratch buffer in d_ws:
//   p0: y  -> ws   (step=1 so blended==y, lap(y))
//   p1: ws -> out
//   p2: out-> ws
//   p3: ws -> out
// ---------------------------------------------------------------------------
extern "C" void kernel_launch(void* const* d_in, const int* in_sizes, int n_in,
                              void* d_out, int out_size, void* d_ws, size_t ws_size,
                              hipStream_t stream) {
  const float* y = (const float*)d_in[0];
  const float* h = (const float*)d_in[1];
  const float* s = (const float*)d_in[2];
  float* out = (float*)d_out;
  float* ws0 = (float*)d_ws;

  const int B = in_sizes[0] / (HH * WW);
  dim3 grid(WW / TILE, HH / TILE, B);
  dim3 blk(256);

  dsgnlm_pass<<<grid, blk, 0, stream>>>(y,   y, h, 0, s, -1, ws0);
  dsgnlm_pass<<<grid, blk, 0, stream>>>(ws0, y, h, 1, s,  0, out);
  dsgnlm_pass<<<grid, blk, 0, stream>>>(out, y, h, 2, s,  1, ws0);
  dsgnlm_pass<<<grid, blk, 0, stream>>>(ws0, y, h, 3, s,  2, out);
}